// Group_54571854463377
// MI455X (gfx1250) — compile-verified
//
#include <hip/hip_runtime.h>

typedef __attribute__((ext_vector_type(2))) float        v2f;
typedef __attribute__((ext_vector_type(8))) float        v8f;
typedef __attribute__((ext_vector_type(4))) unsigned int u32x4;
typedef __attribute__((ext_vector_type(8))) int          i32x8;
typedef __attribute__((ext_vector_type(4))) int          i32x4;

#define B_SZ   32
#define N_PTS  8192
#define N_GRP  512
#define GRP_SZ 32
#define KNN_WAVES 4

#if __has_builtin(__builtin_amdgcn_tensor_load_to_lds) && \
    __has_builtin(__builtin_amdgcn_s_wait_tensorcnt)
#define HAVE_TDM 1
#else
#define HAVE_TDM 0
#endif

// ---------------------------------------------------------------------------
// Tensor Data Mover: 1-D DMA of `ndw` dwords (contiguous) global -> LDS.
// D# group0: count=1 | lds_addr | global_addr[56:0] | type=2.
// D# group1: data_size=4B; tensor_dim0 = tile_dim0 = ndw (1-D tile,
// tile_dim1=0 unused); tensor_dim1=1; stride0=ndw. Groups 2/3 zero (<=2D).
// LDS byte offset = low 32 bits of the generic shared-aperture pointer.
// NOTE: this toolchain exposes the 6-arg builtin
//   (u32x4 g0, i32x8 g1, i32x4, i32x4, i32x8, imm cpol)
// ---------------------------------------------------------------------------
#if HAVE_TDM
__device__ __forceinline__ void tdm_load_dwords(const float* __restrict__ gsrc,
                                                float* ldst, unsigned ndw) {
    const unsigned long long ga = (unsigned long long)(const void*)gsrc;
    const unsigned lo = (unsigned)(unsigned long long)(void*)ldst;
    u32x4 g0;
    g0[0] = 1u;                                             // count=1 (user D#)
    g0[1] = lo;                                             // lds_addr
    g0[2] = (unsigned)(ga & 0xffffffffull);                 // global_addr[31:0]
    g0[3] = (unsigned)((ga >> 32) & 0x01ffffffull)          // global_addr[56:32]
            | (2u << 30);                                   // type=2 ("image")
    i32x8 g1;
    g1[0] = (int)(2u << 16);                                // data_size=4B
    g1[1] = (int)((ndw & 0xffffu) << 16);                   // tensor_dim0[15:0]
    g1[2] = (int)(((ndw >> 16) & 0xffffu) | (1u << 16));    // dim0[31:16]|dim1=1
    g1[3] = (int)((ndw & 0xffffu) << 16);                   // tile_dim0 = ndw
    g1[4] = 0;                                              // tile_dim1/2 unused
    g1[5] = (int)ndw;                                       // dim0_stride[31:0]
    g1[6] = 0;
    g1[7] = 0;
    i32x4 gz4 = {0, 0, 0, 0};
    i32x8 gz8 = {0, 0, 0, 0, 0, 0, 0, 0};
    __builtin_amdgcn_tensor_load_to_lds(g0, g1, gz4, gz4, gz8, 0);
}
#endif

// ---------------------------------------------------------------------------
// Kernel 1: Farthest-point sampling. One workgroup per batch; interleaved
// point cloud (96KB, TDM-loaded) + running min distance (32KB) in LDS.
// 512 sequential steps: distance update + block argmax (tie -> lower index).
// ---------------------------------------------------------------------------
__global__ __launch_bounds__(256) void fps_kernel(const float* __restrict__ xyz,
                                                  float* __restrict__ centers)
{
    extern __shared__ float smem[];
    float* sp = smem;                 // [3*N_PTS] interleaved xyz
    float* sd = smem + 3 * N_PTS;     // [N_PTS]
    __shared__ float rv[8];
    __shared__ int   ri[8];
    __shared__ int   sfar;

    const int b    = blockIdx.x;
    const int tid  = threadIdx.x;
    const int lane = tid & 31;
    const int wave = tid >> 5;
    const float* p = xyz + (size_t)b * N_PTS * 3;

#if HAVE_TDM
    if (wave == 0) {                              // uniform branch: one issue
        tdm_load_dwords(p, sp, 3 * N_PTS);
        __builtin_amdgcn_s_wait_tensorcnt(0);
    }
#else
    for (int i = tid; i < 3 * N_PTS; i += 256) sp[i] = p[i];
#endif
    for (int i = tid; i < N_PTS; i += 256) sd[i] = 1e10f;
    if (tid == 0) sfar = 0;                       // deterministic start at 0
    __syncthreads();

    for (int s = 0; s < N_GRP; ++s) {
        const int far = sfar;
        const float cx = sp[3*far+0], cy = sp[3*far+1], cz = sp[3*far+2];
        if (tid == 0) {
            size_t o = ((size_t)b * N_GRP + s) * 3;
            centers[o+0] = cx; centers[o+1] = cy; centers[o+2] = cz;
        }
        float bv = -1.0f;
        int   bi = 0x7fffffff;
        for (int i = tid; i < N_PTS; i += 256) {
            float dx = sp[3*i+0]-cx, dy = sp[3*i+1]-cy, dz = sp[3*i+2]-cz;
            float d  = dx*dx + dy*dy + dz*dz;
            float nd = fminf(sd[i], d);
            sd[i] = nd;
            if (nd > bv) { bv = nd; bi = i; }     // strict >: earliest index
        }
        for (int off = 16; off > 0; off >>= 1) {  // wave32 argmax reduction
            float ov = __shfl_down(bv, off, 32);
            int   oi = __shfl_down(bi, off, 32);
            if (ov > bv || (ov == bv && oi < bi)) { bv = ov; bi = oi; }
        }
        if (lane == 0) { rv[wave] = bv; ri[wave] = bi; }
        __syncthreads();
        if (tid == 0) {
            float v = rv[0]; int ix = ri[0];
            for (int w = 1; w < 8; ++w)
                if (rv[w] > v || (rv[w] == v && ri[w] < ix)) { v = rv[w]; ix = ri[w]; }
            sfar = ix;
        }
        __syncthreads();
    }
}

// ---------------------------------------------------------------------------
// Kernel 2: KNN top-32 + gather. Each wave owns 16 centers. The FULL squared
// distance comes out of one V_WMMA_F32_16X16X4_F32 per 16-point tile:
//   A[m] = (-2cx,-2cy,-2cz, 1),  B[n] = (px,py,pz,|p|^2),  C[m][n] = |c|^2
//   => D[m][n] = |c|^2 + |p|^2 - 2 c.p
// LDS float offsets:
#define OFF_SP   0                                     // [3*N_PTS] interleaved
#define OFF_TOPD (3 * N_PTS)                           // [4][16][32] float
#define OFF_TOPI (OFF_TOPD + KNN_WAVES * 16 * GRP_SZ)  // [4][16][32] int
#define OFF_DC   (OFF_TOPI + KNN_WAVES * 16 * GRP_SZ)  // [4][16][16] float
#define KNN_SMEM_FLOATS (OFF_DC + KNN_WAVES * 16 * 16)
// ---------------------------------------------------------------------------
__global__ __launch_bounds__(128) void knn_kernel(const float* __restrict__ xyz,
                                                  const float* __restrict__ centers,
                                                  float* __restrict__ neigh)
{
    extern __shared__ float smem[];
    float* sp   = smem + OFF_SP;
    float* topD = smem + OFF_TOPD;
    int*   topI = (int*)(smem + OFF_TOPI);
    float* dC   = smem + OFF_DC;

    const int b    = blockIdx.y;
    const int tid  = threadIdx.x;
    const int lane = tid & 31;
    const int wave = tid >> 5;
    const int g0   = blockIdx.x * (KNN_WAVES * 16) + wave * 16;

    const float* p = xyz + (size_t)b * N_PTS * 3;
#if HAVE_TDM
    if (wave == 0) {
        tdm_load_dwords(p, sp, 3 * N_PTS);
        __builtin_amdgcn_s_wait_tensorcnt(0);
    }
#else
    for (int i = tid; i < 3 * N_PTS; i += 128) sp[i] = p[i];
#endif
    for (int i = lane; i < 16 * GRP_SZ; i += 32) {        // init top-32 lists
        topD[wave * 16 * GRP_SZ + i] = 3.0e38f;
        topI[wave * 16 * GRP_SZ + i] = 0;
    }

    // A matrix (ISA 16x4 f32 layout: lanes 0-15 -> K0,K1; lanes 16-31 -> K2,K3)
    const int m = lane & 15;
    const float* cp = centers + ((size_t)b * N_GRP + g0 + m) * 3;
    __builtin_prefetch(cp, 0, 1);                          // global_prefetch_b8
    const float ccx = cp[0], ccy = cp[1], ccz = cp[2];
    v2f A;
    A[0] = (lane < 16) ? -2.0f * ccx : -2.0f * ccz;
    A[1] = (lane < 16) ? -2.0f * ccy : 1.0f;
    // C matrix: C[m][n] = |c_m|^2 ; lane<16 VGPR j <-> M=j, lane>=16 <-> M=j+8
    const int   mbase = (lane < 16) ? 0 : 8;
    const float cn2m  = ccx*ccx + ccy*ccy + ccz*ccz;       // center lane&15
    v8f Cs;
    #pragma unroll
    for (int j = 0; j < 8; ++j) Cs[j] = __shfl(cn2m, mbase + j, 32);
    __syncthreads();

    const int n = lane & 15;
    float thr = 3.0e38f;                                   // cached myD[31]
    #pragma unroll 2
    for (int c = 0; c < N_PTS / 16; ++c) {
        const int pt = c * 16 + n;
        const float px = sp[3*pt+0], py = sp[3*pt+1], pz = sp[3*pt+2];
        const float pn2 = px*px + py*py + pz*pz;
        v2f Bm;                                            // B: K0,K1 / K2,K3
        Bm[0] = (lane < 16) ? px : pz;
        Bm[1] = (lane < 16) ? py : pn2;
        // D = full squared distance tile (16 centers x 16 points)
        v8f D = __builtin_amdgcn_wmma_f32_16x16x4_f32(
            false, A, false, Bm, (short)0, Cs, false, false);
        #pragma unroll
        for (int j = 0; j < 8; ++j)
            dC[(wave * 16 + mbase + j) * 16 + n] = D[j];
        // merge tile into per-center sorted top-32 (lane == center)
        if (lane < 16) {
            float* myD = topD + (wave * 16 + lane) * GRP_SZ;
            int*   myI = topI + (wave * 16 + lane) * GRP_SZ;
            for (int q = 0; q < 16; ++q) {
                float d = dC[(wave * 16 + lane) * 16 + q];
                if (d < thr) {                             // strict <: stable
                    int pos = GRP_SZ - 1;
                    while (pos > 0 && d < myD[pos - 1]) {
                        myD[pos] = myD[pos - 1];
                        myI[pos] = myI[pos - 1];
                        --pos;
                    }
                    myD[pos] = d;
                    myI[pos] = c * 16 + q;
                    thr = myD[GRP_SZ - 1];
                }
            }
        }
    }

    // gather: neighborhood = points[idx] - center ; lane = neighbor k
    for (int mm = 0; mm < 16; ++mm) {
        const int g = g0 + mm;
        const size_t co = ((size_t)b * N_GRP + g) * 3;
        const float gx = centers[co+0], gy = centers[co+1], gz = centers[co+2];
        const int idx = topI[(wave * 16 + mm) * GRP_SZ + lane];
        const size_t o = (((size_t)b * N_GRP + g) * GRP_SZ + lane) * 3;
        neigh[o+0] = sp[3*idx+0] - gx;
        neigh[o+1] = sp[3*idx+1] - gy;
        neigh[o+2] = sp[3*idx+2] - gz;
    }
}

// ---------------------------------------------------------------------------
extern "C" void kernel_launch(void* const* d_in, const int* in_sizes, int n_in,
                              void* d_out, int out_size, void* d_ws, size_t ws_size,
                              hipStream_t stream) {
    (void)in_sizes; (void)n_in; (void)out_size; (void)d_ws; (void)ws_size;
    const float* xyz = (const float*)d_in[0];
    float* out     = (float*)d_out;
    float* neigh   = out;                                       // [B,G,32,3]
    float* centers = out + (size_t)B_SZ * N_GRP * GRP_SZ * 3;   // [B,G,3]

    fps_kernel<<<dim3(B_SZ), dim3(256), 4 * N_PTS * sizeof(float), stream>>>(
        xyz, centers);

    knn_kernel<<<dim3(N_GRP / (KNN_WAVES * 16), B_SZ), dim3(128),
                 KNN_SMEM_FLOATS * sizeof(float), stream>>>(
        xyz, centers, neigh);
}